// DetectBox_60962765799883
// MI455X (gfx1250) — compile-verified
//
#include <hip/hip_runtime.h>

#define NUM_B 16
#define NUM_A 32768
#define NUM_C 81
#define MAX_TOTAL 100
#define MAX_PER_CLASS 100
#define SCORE_THR 0.5f
#define IOU_THR 0.3f

typedef float v2f __attribute__((ext_vector_type(2)));
typedef float v8f __attribute__((ext_vector_type(8)));

// ---------------------------------------------------------------------------
// Kernel A: per-anchor max-softmax score + argmax class + box decode.
// Block = 256 threads = 8 waves; block stages its 128x81 logits tile into LDS
// with GLOBAL_LOAD_ASYNC_TO_LDS_B128 (ASYNCcnt). One wave = 16 anchors; the
// softmax denominator (row-sum of exp) is computed with V_WMMA_F32_16X16X4_F32
// against an all-ones B matrix => exact f32 row sums.
// ---------------------------------------------------------------------------
__global__ __launch_bounds__(256) void score_decode_kernel(
    const float* __restrict__ deltas,   // [B,A,4]
    const float* __restrict__ logits,   // [B,A,81]
    const float* __restrict__ anchors,  // [A,4]
    float* __restrict__ gscore,         // [B,A]  live score or -1
    int*   __restrict__ gid,            // [B,A]  argmax class
    float* __restrict__ gbox)           // [B,A,4] decoded boxes
{
    const int b    = blockIdx.y;
    const int tid  = threadIdx.x;
    const int wave = tid >> 5;
    const int lane = tid & 31;
    const int half = lane >> 4;     // 0: K even pair, 1: K odd pair (A-frag layout)
    const int m    = lane & 15;     // anchor row within the 16-anchor tile
    const int lrow = wave * 16 + m; // 0..127 local anchor within block
    const int a    = blockIdx.x * 128 + lrow;

    __shared__ float tile[128 * NUM_C];   // 41472 B logits tile

    // ---- async coalesced copy: global logits tile -> LDS (16B per lane chunk)
    {
        const char* gbase = (const char*)(logits
                            + ((long)b * NUM_A + (long)blockIdx.x * 128) * NUM_C);
        const unsigned lds0 = (unsigned)(uintptr_t)(&tile[0]);
        const int nchunks = (128 * NUM_C * 4) / 16;   // 2592 16-byte chunks
        for (int k = tid; k < nchunks; k += 256) {
            const unsigned ldsa = lds0 + (unsigned)(k * 16);
            const char*    ga   = gbase + (long)k * 16;
            asm volatile("global_load_async_to_lds_b128 %0, %1, off"
                         :: "v"(ldsa), "v"(ga)
                         : "memory");
        }
        asm volatile("s_wait_asynccnt 0x0" ::: "memory");
        __syncthreads();
    }

    const float* row = &tile[lrow * NUM_C];

    // ---- pass 1: max/argmax over 81 logits (2 per lane per 4-class chunk)
    float bestv = -3.402823466e+38f;
    int   besti = 0;
#pragma unroll
    for (int t = 0; t < 20; ++t) {
        const int c0 = 4 * t + 2 * half;
        const float x0 = row[c0];
        const float x1 = row[c0 + 1];
        if (x0 > bestv) { bestv = x0; besti = c0; }
        if (x1 > bestv) { bestv = x1; besti = c0 + 1; }
    }
    {
        const float x = row[80];
        if (x > bestv) { bestv = x; besti = 80; }
    }
    // combine the two half-lanes holding this anchor (lowest-index tie-break)
    {
        const float ov = __shfl_xor(bestv, 16, 32);
        const int   oi = __shfl_xor(besti, 16, 32);
        if (ov > bestv || (ov == bestv && oi < besti)) { bestv = ov; besti = oi; }
    }

    // ---- pass 2: sumexp via chained WMMA f32 16x16x4, B = ones => row sums
    v8f acc = {0.f, 0.f, 0.f, 0.f, 0.f, 0.f, 0.f, 0.f};
    v2f ones; ones.x = 1.0f; ones.y = 1.0f;
#pragma unroll
    for (int t = 0; t < 20; ++t) {
        const int c0 = 4 * t + 2 * half;
        v2f afrag;
        afrag.x = __expf(row[c0]     - bestv);
        afrag.y = __expf(row[c0 + 1] - bestv);
        acc = __builtin_amdgcn_wmma_f32_16x16x4_f32(
                false, afrag, false, ones, (short)0, acc, false, false);
    }
    {   // class 80 only (pad K=81..83 with zeros; only half 0 contributes)
        v2f afrag;
        afrag.x = (half == 0) ? __expf(row[80] - bestv) : 0.0f;
        afrag.y = 0.0f;
        acc = __builtin_amdgcn_wmma_f32_16x16x4_f32(
                false, afrag, false, ones, (short)0, acc, false, false);
    }

    // ---- extract row m from D fragment. VGPR v of lane L holds
    // D[v + 8*(L>=16)][L&15]; all columns equal (B = ones).
    const int r = lane & 7;
    const float a0 = acc[0], a1 = acc[1], a2 = acc[2], a3 = acc[3];
    const float a4 = acc[4], a5 = acc[5], a6 = acc[6], a7 = acc[7];
    const float m01 = (r & 1) ? a1 : a0;
    const float m23 = (r & 1) ? a3 : a2;
    const float m45 = (r & 1) ? a5 : a4;
    const float m67 = (r & 1) ? a7 : a6;
    const float m03 = (r & 2) ? m23 : m01;
    const float m47 = (r & 2) ? m67 : m45;
    const float msel = (r & 4) ? m47 : m03;        // lane L holds row (L&7)+8*half
    const float sumexp = __shfl(msel, m + (m & 8), 32);  // fetch row m

    const float score = 1.0f / sumexp;             // max softmax = exp(0)/sumexp
    const int valid = (besti > 0) && (score >= SCORE_THR);

    if (half == 0) {
        const long idx = (long)b * NUM_A + a;
        gscore[idx] = valid ? score : -1.0f;
        gid[idx]    = besti;

        // box decode
        const float* an = anchors + (long)a * 4;
        const float* de = deltas + idx * 4;
        const float ay1 = an[0], ax1 = an[1], ay2 = an[2], ax2 = an[3];
        float h = ay2 - ay1, w = ax2 - ax1;
        float cy = (ay2 + ay1) * 0.5f, cx = (ax2 + ax1) * 0.5f;
        cy += de[0] * 0.1f * h;
        cx += de[1] * 0.1f * w;
        h *= __expf(de[2] * 0.2f);
        w *= __expf(de[3] * 0.2f);
        float* bo = gbox + idx * 4;
        bo[0] = cy - 0.5f * h;
        bo[1] = cx - 0.5f * w;
        bo[2] = cy + 0.5f * h;
        bo[3] = cx + 0.5f * w;
    }
}

// ---------------------------------------------------------------------------
// Kernel B: 100-round greedy class-aware NMS, one 1024-thread block per image.
// LDS: per-anchor kill flags (bytes, thread-owned), argmax reduction arrays,
// per-class counts.
// ---------------------------------------------------------------------------
__global__ __launch_bounds__(1024) void nms_kernel(
    const float* __restrict__ gscore,
    const int*   __restrict__ gid,
    const float* __restrict__ gbox,
    float* __restrict__ out)
{
    const int  b    = blockIdx.x;
    const int  tid  = threadIdx.x;
    const long base = (long)b * NUM_A;

    __shared__ unsigned char killed[NUM_A];   // 32 KB
    __shared__ float rv[1024];
    __shared__ int   ri[1024];
    __shared__ int   counts[NUM_C];
    __shared__ float bb[4];
    __shared__ int   bc, bok, bfull;

    for (int j = tid; j < NUM_A; j += 1024) killed[j] = 0;
    if (tid < NUM_C) counts[tid] = 0;
    __syncthreads();

    float* out_boxes  = out;                                // [B,100,5]
    float* out_scores = out + (long)NUM_B * MAX_TOTAL * 5;  // [B,100,2]
    float* out_ids    = out + (long)NUM_B * MAX_TOTAL * 7;  // [B,100,2]

    for (int round = 0; round < MAX_TOTAL; ++round) {
        // --- argmax over live entries (killed excluded; -1 entries inert)
        float best = -3.402823466e+38f;
        int   bidx = 0;
        for (int j = tid; j < NUM_A; j += 1024) {
            if (!killed[j]) {
                const float s = gscore[base + j];
                if (s > best) { best = s; bidx = j; }   // strict > keeps lowest idx
            }
        }
        rv[tid] = best; ri[tid] = bidx;
        __syncthreads();
        for (int off = 512; off > 0; off >>= 1) {
            if (tid < off) {
                const float v2 = rv[tid + off]; const int i2 = ri[tid + off];
                const float v1 = rv[tid];       const int i1 = ri[tid];
                if (v2 > v1 || (v2 == v1 && i2 < i1)) { rv[tid] = v2; ri[tid] = i2; }
            }
            __syncthreads();
        }

        if (tid == 0) {
            const float s = rv[0];
            const int   i = ri[0];
            const int   ok = (s > 0.0f);
            int c = 0, full = 0;
            float b0 = 0.f, b1 = 0.f, b2 = 0.f, b3 = 0.f;
            if (ok) {
                c = gid[base + i];
                const int cnt = counts[c] + 1;
                counts[c] = cnt;
                full = (cnt >= MAX_PER_CLASS);
                const float* bp = gbox + (base + i) * 4;
                b0 = bp[0]; b1 = bp[1]; b2 = bp[2]; b3 = bp[3];
            }
            bb[0] = b0; bb[1] = b1; bb[2] = b2; bb[3] = b3;
            bc = c; bok = ok; bfull = full;

            const float flag = ok ? 1.0f : 0.0f;
            float* ob = out_boxes + ((long)b * MAX_TOTAL + round) * 5;
            ob[0] = b0 * flag; ob[1] = b1 * flag; ob[2] = b2 * flag; ob[3] = b3 * flag;
            ob[4] = flag;
            float* os = out_scores + ((long)b * MAX_TOTAL + round) * 2;
            os[0] = ok ? s : 0.0f; os[1] = flag;
            float* oi = out_ids + ((long)b * MAX_TOTAL + round) * 2;
            oi[0] = ok ? (float)c : 0.0f; oi[1] = flag;
        }
        __syncthreads();

        // --- suppression pass
        if (bok) {
            const float sy1 = bb[0], sx1 = bb[1], sy2 = bb[2], sx2 = bb[3];
            const float area1 = (sy2 - sy1) * (sx2 - sx1);
            const int c = bc, full = bfull;
            for (int j = tid; j < NUM_A; j += 1024) {
                if (killed[j]) continue;
                if (gid[base + j] != c) continue;
                if (full) { killed[j] = 1; continue; }
                const float* bp = gbox + (base + j) * 4;
                const float y1 = bp[0], x1 = bp[1], y2 = bp[2], x2 = bp[3];
                const float yy1 = fmaxf(sy1, y1), xx1 = fmaxf(sx1, x1);
                const float yy2 = fminf(sy2, y2), xx2 = fminf(sx2, x2);
                const float inter = fmaxf(yy2 - yy1, 0.0f) * fmaxf(xx2 - xx1, 0.0f);
                const float area2 = (y2 - y1) * (x2 - x1);
                const float iou = inter / (area1 + area2 - inter + 1e-8f);
                if (iou > IOU_THR) killed[j] = 1;
            }
        }
        __syncthreads();
    }
}

// ---------------------------------------------------------------------------
extern "C" void kernel_launch(void* const* d_in, const int* in_sizes, int n_in,
                              void* d_out, int out_size, void* d_ws, size_t ws_size,
                              hipStream_t stream) {
    (void)in_sizes; (void)n_in; (void)out_size; (void)ws_size;
    const float* deltas  = (const float*)d_in[0];
    const float* logits  = (const float*)d_in[1];
    const float* anchors = (const float*)d_in[2];
    float* out = (float*)d_out;

    char* ws = (char*)d_ws;
    const long BA = (long)NUM_B * NUM_A;
    float* gscore = (float*)ws;                       // BA floats
    int*   gid    = (int*)(ws + BA * 4);              // BA ints
    float* gbox   = (float*)(ws + BA * 8);            // BA*4 floats (total 12.6 MB)

    dim3 gridA(NUM_A / 128, NUM_B);                   // wave = 16 anchors, 8 waves/block
    score_decode_kernel<<<gridA, 256, 0, stream>>>(deltas, logits, anchors,
                                                   gscore, gid, gbox);
    nms_kernel<<<NUM_B, 1024, 0, stream>>>(gscore, gid, gbox, out);
}